// TactileSpikeEncoder_44092134260936
// MI455X (gfx1250) — compile-verified
//
#include <hip/hip_runtime.h>
#include <hip/hip_bf16.h>
#include <stdint.h>

// ---------------- problem constants (from reference) ----------------
#define B_DIM   32
#define F_DIM   64
#define BF      (B_DIM * F_DIM)          // 2048 rows
#define N_NEU   256
#define T_STEPS 100
#define N_ELEMS (BF * N_NEU)             // 524288 flattened [B,F,N]
#define HALF_E  (N_ELEMS / 2)            // 262144 : JAX threefry counter split
#define SPIKE_SCALE 0.1f                 // SPIKE_RATE/1000
#define REFRACTORY  2.0f

typedef __attribute__((ext_vector_type(2))) float v2f;
typedef __attribute__((ext_vector_type(8))) float v8f;

// ---------------- Threefry-2x32-20 (JAX-compatible) ----------------
__device__ __forceinline__ uint32_t rotl32(uint32_t v, int s) {
    return (v << s) | (v >> (32 - s));
}

__device__ __forceinline__ void threefry2x32(uint32_t k0, uint32_t k1,
                                             uint32_t c0, uint32_t c1,
                                             uint32_t& o0, uint32_t& o1) {
    uint32_t ks2 = 0x1BD11BDAu ^ k0 ^ k1;
    uint32_t x0 = c0 + k0;
    uint32_t x1 = c1 + k1;
#define TFR(r) { x0 += x1; x1 = rotl32(x1, (r)); x1 ^= x0; }
    TFR(13) TFR(15) TFR(26) TFR(6)
    x0 += k1;  x1 += ks2 + 1u;
    TFR(17) TFR(29) TFR(16) TFR(24)
    x0 += ks2; x1 += k0 + 2u;
    TFR(13) TFR(15) TFR(26) TFR(6)
    x0 += k0;  x1 += k1 + 3u;
    TFR(17) TFR(29) TFR(16) TFR(24)
    x0 += k1;  x1 += ks2 + 4u;
    TFR(13) TFR(15) TFR(26) TFR(6)
    x0 += ks2; x1 += k0 + 5u;
#undef TFR
    o0 = x0; o1 = x1;
}

// JAX float32 uniform in [0,1): (bits>>9)|0x3f800000 as float, minus 1.
__device__ __forceinline__ float u01(uint32_t bits) {
    return __uint_as_float((bits >> 9) | 0x3F800000u) - 1.0f;
}

// ---------------- Phase 0: spike_prob via V_WMMA_F32_16X16X4_F32 ----------
// (x-c)^2 / (2 w^2) = [x^2, x, 1] . [inv, -2 c inv, c^2 inv]   (inv = 1/(2w^2))
// A: 16x4 f32 (rows = bf tile), B: 4x16 f32 (cols = neuron tile), C: 16x16 f32.
// One wave32 per 16x16 tile. EXEC all ones; only ternaries, no branches.
__global__ void probKernel(const float* __restrict__ x,
                           const float* __restrict__ centers,
                           const float* __restrict__ widths,
                           const float* __restrict__ adapt,
                           float* __restrict__ p) {
    const int lane = threadIdx.x;        // 0..31
    const int m0 = blockIdx.x * 16;      // bf tile   (128 tiles)
    const int n0 = blockIdx.y * 16;      // neuron tile (16 tiles)
    const int sub = lane & 15;
    const bool lo = lane < 16;

    const float xv  = x[m0 + sub];
    const int   n   = n0 + sub;
    const float c   = centers[n];
    const float w   = widths[n];
    const float inv = 1.0f / (2.0f * w * w);

    // A layout (32-bit 16x4): lanes 0-15 hold K=0,1 ; lanes 16-31 hold K=2,3.
    v2f a;
    a.x = lo ? xv * xv : 1.0f;           // K0 : x^2     | K2 : 1
    a.y = lo ? xv      : 0.0f;           // K1 : x       | K3 : 0 (pad)
    // B layout (4x16): lanes 0-15 K=0,1 for column N=lane; lanes 16-31 K=2,3.
    v2f b;
    b.x = lo ? inv                 : c * c * inv;   // K0 | K2
    b.y = lo ? (-2.0f * c * inv)   : 0.0f;          // K1 | K3

    v8f acc = {};
    acc = __builtin_amdgcn_wmma_f32_16x16x4_f32(
        /*neg_a=*/false, a, /*neg_b=*/false, b,
        /*c_mod=*/(short)0, acc, /*reuse_a=*/false, /*reuse_b=*/false);

    // D layout: VGPR j -> row M = j (lanes 0-15) or j+8 (lanes 16-31), col = sub.
    const float scale = adapt[n] * SPIKE_SCALE;
    const int mbase = m0 + (lo ? 0 : 8);
#pragma unroll
    for (int j = 0; j < 8; ++j) {
        float pv = expf(-acc[j]) * scale;
        p[(mbase + j) * N_NEU + n] = pv;
    }
}

// ---------------- Phase 1: per-timestep fold_in keys ----------------
__global__ void fkeyKernel(const int* __restrict__ seedp,
                           uint32_t* __restrict__ fkeys) {
    const int t = threadIdx.x;
    if (t >= T_STEPS) return;
    const long long s = (long long)seedp[0];
    const uint32_t k0 = (uint32_t)(s >> 32);
    const uint32_t k1 = (uint32_t)s;
    uint32_t f0, f1;
    threefry2x32(k0, k1, 0u, (uint32_t)t, f0, f1);   // fold_in(key, t)
    fkeys[2 * t]     = f0;
    fkeys[2 * t + 1] = f1;
}

// ---------------- Phase 2: any raw spike per (neuron, t) ----------------
// Counter pair (e, e+HALF) shares the same neuron (HALF = 1024*256), so one
// threefry covers two samples of the same reduction.
__global__ void anyRawKernel(const float* __restrict__ p,
                             const uint32_t* __restrict__ fkeys,
                             uint32_t* __restrict__ any_raw) {
    const int n = blockIdx.x;            // 0..255
    const int t = blockIdx.y;            // 0..99
    __shared__ uint32_t flag;
    if (threadIdx.x == 0) flag = 0u;
    __syncthreads();

    const uint32_t k0 = fkeys[2 * t];
    const uint32_t k1 = fkeys[2 * t + 1];
    uint32_t local = 0u;
#pragma unroll
    for (int i = 0; i < 4; ++i) {
        const unsigned bf = threadIdx.x + i * 256;   // 0..1023
        const unsigned e  = bf * N_NEU + n;
        uint32_t o0, o1;
        threefry2x32(k0, k1, e, e + HALF_E, o0, o1);
        local |= (u01(o0) < p[e])          ? 1u : 0u;
        local |= (u01(o1) < p[e + HALF_E]) ? 1u : 0u;
    }
    if (local) atomicOr(&flag, 1u);
    __syncthreads();
    if (threadIdx.x == 0) any_raw[t * N_NEU + n] = flag;
}

// ---------------- Phase 3: sequential refractory recurrence --------------
__global__ void refKernel(const float* __restrict__ refr_in,
                          const uint32_t* __restrict__ any_raw,
                          float* __restrict__ can) {
    const int n = threadIdx.x;           // 256 threads, 1 per neuron
    float ref = refr_in[n];
#pragma unroll 4
    for (int t = 0; t < T_STEPS; ++t) {
        const float cs = (ref == 0.0f) ? 1.0f : 0.0f;
        const bool occurred = (any_raw[t * N_NEU + n] != 0u) && (cs != 0.0f);
        can[t * N_NEU + n] = cs;
        ref = fmaxf(ref - 1.0f, 0.0f) + (occurred ? REFRACTORY : 0.0f);
    }
}

// ---------------- Phase 4: gated spike writeout (bandwidth bound) --------
// One thread per (counter-pair, t): 1 threefry -> 2 output elements.
// out flat index = e*T + t (matches [B,F,N,T] layout).
__global__ void spikesKernel(const float* __restrict__ p,
                             const float* __restrict__ can,
                             const uint32_t* __restrict__ fkeys,
                             float* __restrict__ out) {
    const unsigned gid = blockIdx.x * blockDim.x + threadIdx.x;
    const unsigned total = (unsigned)HALF_E * T_STEPS;   // 26,214,400
    if (gid >= total) return;
    const unsigned pi = gid / T_STEPS;       // element index < HALF_E
    const unsigned t  = gid - pi * T_STEPS;
    const unsigned n  = pi & (N_NEU - 1);    // pair shares neuron
    const unsigned e1 = pi + HALF_E;

    uint32_t o0, o1;
    threefry2x32(fkeys[2 * t], fkeys[2 * t + 1], pi, e1, o0, o1);

    const float canv = can[t * N_NEU + n];
    const bool g = (canv != 0.0f);
    const float r0 = u01(o0);
    const float r1 = u01(o1);
    out[(size_t)pi * T_STEPS + t] = (g && (r0 < p[pi])) ? 1.0f : 0.0f;
    out[(size_t)e1 * T_STEPS + t] = (g && (r1 < p[e1])) ? 1.0f : 0.0f;
}

// ---------------- launcher ----------------
extern "C" void kernel_launch(void* const* d_in, const int* in_sizes, int n_in,
                              void* d_out, int out_size, void* d_ws, size_t ws_size,
                              hipStream_t stream) {
    const float* x       = (const float*)d_in[0];   // [32,64]
    const float* centers = (const float*)d_in[1];   // [256]
    const float* widths  = (const float*)d_in[2];   // [256]
    const float* adapt   = (const float*)d_in[3];   // [256]
    const float* refr    = (const float*)d_in[4];   // [256]
    const int*   seed    = (const int*)d_in[5];     // scalar

    // workspace layout
    char* ws = (char*)d_ws;
    float*    p       = (float*)ws;                                   // 2048*256 f32 (2 MB)
    size_t    off     = (size_t)N_ELEMS * sizeof(float);
    uint32_t* any_raw = (uint32_t*)(ws + off);                        // 100*256 u32
    off += (size_t)T_STEPS * N_NEU * sizeof(uint32_t);
    float*    can     = (float*)(ws + off);                           // 100*256 f32
    off += (size_t)T_STEPS * N_NEU * sizeof(float);
    uint32_t* fkeys   = (uint32_t*)(ws + off);                        // 100*2 u32

    // Phase 0: spike probabilities via WMMA (128 x 16 tiles, one wave32 each)
    probKernel<<<dim3(BF / 16, N_NEU / 16), 32, 0, stream>>>(x, centers, widths, adapt, p);
    // Phase 1: per-timestep fold_in keys
    fkeyKernel<<<1, 128, 0, stream>>>(seed, fkeys);
    // Phase 2: any-raw-spike reduction per (neuron, timestep)
    anyRawKernel<<<dim3(N_NEU, T_STEPS), 256, 0, stream>>>(p, fkeys, any_raw);
    // Phase 3: sequential refractory recurrence -> can_spike table
    refKernel<<<1, N_NEU, 0, stream>>>(refr, any_raw, can);
    // Phase 4: gated spike writeout
    const unsigned total = (unsigned)HALF_E * T_STEPS;
    spikesKernel<<<(total + 255) / 256, 256, 0, stream>>>(p, can, fkeys, (float*)d_out);
}